// ClusterNet_20547123544890
// MI455X (gfx1250) — compile-verified
//
#include <hip/hip_runtime.h>
#include <math.h>

// ---------------------------------------------------------------------------
// Soft spherical clustering (11 fused iterations) for MI455X / gfx1250.
// Both GEMMs (cos = data@mn^T and cluster_mean += r^T@data) run on
// V_WMMA_F32_16X16X4_F32 in full fp32. K-slot<->data mappings are chosen so
// that (a) GEMM1 operands stream as contiguous b128 LDS loads feeding two
// WMMAs each, and (b) the GEMM2 A-fragment (r^T) is lane-resident (no
// cross-lane traffic). Per-row norm scales are SGPR-broadcast via
// readfirstlane'd addresses. Target: 11 x 256MB streamed = ~120us at
// 23.3 TB/s, with instruction issue ~3x under the memory roofline.
// ---------------------------------------------------------------------------

typedef __attribute__((ext_vector_type(2))) float v2f;
typedef __attribute__((ext_vector_type(4))) float v4f;
typedef __attribute__((ext_vector_type(8))) float v8f;

#define NN      500000
#define DD      128
#define KK      16
#define NTILES  (NN / 16)     // 31250 tiles of 16 rows
#define TSTRIDE 132           // padded LDS row stride (floats)
#define TEMP    0.5f
#define EPSV    1e-8f
#define NITER   11            // 1 warm-up + 10

__device__ __forceinline__ float sum16(float v) {
  v += __shfl_xor(v, 1, 32);
  v += __shfl_xor(v, 2, 32);
  v += __shfl_xor(v, 4, 32);
  v += __shfl_xor(v, 8, 32);
  return v;
}

// --------------------------------------------------------------------------
// Kernel 1: per-row scales. data = e/(||e||+eps)  => s1 = 1/(||e||+eps)
//           dn = data/||data|| = e/||e||          => rscale = (||e||+eps)/||e||
// One wave32 per row (128 floats = one float4 per lane).
// --------------------------------------------------------------------------
__global__ void rownorm_kernel(const float* __restrict__ emb,
                               float* __restrict__ s1,
                               float* __restrict__ rsc) {
  int gw = (int)((blockIdx.x * blockDim.x + threadIdx.x) >> 5);
  int lane = threadIdx.x & 31;
  if (gw >= NN) return;
  const float4 v = ((const float4*)(emb + (size_t)gw * DD))[lane];
  float ss = v.x * v.x + v.y * v.y + v.z * v.z + v.w * v.w;
  ss += __shfl_xor(ss, 1, 32);
  ss += __shfl_xor(ss, 2, 32);
  ss += __shfl_xor(ss, 4, 32);
  ss += __shfl_xor(ss, 8, 32);
  ss += __shfl_xor(ss, 16, 32);
  float n = sqrtf(ss);
  if (lane == 0) {
    s1[gw]  = 1.0f / (n + EPSV);
    rsc[gw] = (n + EPSV) / n;
  }
}

// --------------------------------------------------------------------------
// Kernel 2: mn = row_normalize(row_normalize(mu)); zero accumulators.
// Single block, 256 threads (8 waves); wave w handles rows w and w+8.
// --------------------------------------------------------------------------
__global__ void muprep_kernel(const float* __restrict__ mu_in,
                              float* __restrict__ mn,
                              float* __restrict__ cmean,
                              float* __restrict__ cr) {
  int tid = threadIdx.x;
  int lane = tid & 31;
  int w = tid >> 5;
  for (int r = w; r < KK; r += 8) {
    const float4 v = ((const float4*)(mu_in + r * DD))[lane];
    float ss = v.x * v.x + v.y * v.y + v.z * v.z + v.w * v.w;
    ss += __shfl_xor(ss, 1, 32);  ss += __shfl_xor(ss, 2, 32);
    ss += __shfl_xor(ss, 4, 32);  ss += __shfl_xor(ss, 8, 32);
    ss += __shfl_xor(ss, 16, 32);
    float inv1 = 1.0f / sqrtf(ss);
    float ax = v.x * inv1, ay = v.y * inv1, az = v.z * inv1, aw = v.w * inv1;
    // second normalize (idempotent up to fp rounding, matches reference)
    float s2 = ax * ax + ay * ay + az * az + aw * aw;
    s2 += __shfl_xor(s2, 1, 32);  s2 += __shfl_xor(s2, 2, 32);
    s2 += __shfl_xor(s2, 4, 32);  s2 += __shfl_xor(s2, 8, 32);
    s2 += __shfl_xor(s2, 16, 32);
    float inv2 = 1.0f / sqrtf(s2);
    float4 o;
    o.x = ax * inv2; o.y = ay * inv2; o.z = az * inv2; o.w = aw * inv2;
    ((float4*)(mn + r * DD))[lane] = o;
  }
  for (int i = tid; i < KK * DD; i += 256) cmean[i] = 0.0f;
  if (tid < KK) cr[tid] = 0.0f;
}

// --------------------------------------------------------------------------
// Kernel 3 (main): fused cos -> softmax -> cluster_r / cluster_mean.
// 128 threads = 4 wave32 per block; each wave grid-strides over 16-row tiles.
// cluster_mean partials live in 64 VGPRs per wave across the whole loop.
// --------------------------------------------------------------------------
__global__ void __launch_bounds__(128)
cluster_main(const float* __restrict__ emb, const float* __restrict__ s1,
             const float* __restrict__ rsc, const float* __restrict__ mn,
             float* __restrict__ cmean, float* __restrict__ cr,
             float* __restrict__ r_out, int write_r) {
  __shared__ __align__(16) float mn_lds[KK * TSTRIDE];
  __shared__ __align__(16) float tile[4][16 * TSTRIDE];
  __shared__ float cm_lds[KK * DD];
  __shared__ float cr_lds[KK];

  const int tid  = threadIdx.x;
  const int lane = tid & 31;
  const int wid  = tid >> 5;
  const int lcol = lane & 15;
  const int h2   = (lane >> 4) & 1;

  // stage mn with padded stride; zero block accumulators
  for (int i = tid; i < KK * DD; i += 128) {
    mn_lds[(i >> 7) * TSTRIDE + (i & 127)] = mn[i];
    cm_lds[i] = 0.0f;
  }
  if (tid < KK) cr_lds[tid] = 0.0f;
  __syncthreads();

  float* tl = &tile[wid][0];
  v8f acc[8] = {};          // cluster_mean partial: M=cluster(jj+8h), N=col(d0*16+lcol)
  float crv = 0.0f;         // cluster_r partial for cluster lcol

  // wave-uniform tile index, forced into SGPRs so per-row scale reads become
  // scalar loads (broadcast) instead of ds_bpermute shuffles.
  const int gw = __builtin_amdgcn_readfirstlane(blockIdx.x * 4 + wid);
  const int nw = gridDim.x * 4;

  for (int t = gw; t < NTILES; t += nw) {
    const size_t trow = (size_t)t * 16;
    const float* tbase = emb + trow * DD;
    const float* s1b = s1 + trow;    // uniform base -> s_load per row
    const float* rb  = rsc + trow;

    if (t + nw < NTILES)
      __builtin_prefetch(emb + (size_t)(t + nw) * 16 * DD, 0, 1);

    // ---- stage tile into LDS, pre-scaled by s1 (=> "data" rows) ----------
#pragma unroll
    for (int i = 0; i < 16; ++i) {
      float4 v = ((const float4*)(tbase + i * DD))[lane];
      float si = s1b[i];                     // SGPR broadcast
      float4 o;
      o.x = v.x * si; o.y = v.y * si; o.z = v.z * si; o.w = v.w * si;
      *(float4*)(tl + i * TSTRIDE + lane * 4) = o;
    }

    // ---- GEMM1: cos = data_tile @ mn^T (fp32 WMMA, K=4) ------------------
    // Slot map: dim(dc, s) = 64*(s>>1) + 2*dc + (s&1). Each lane streams a
    // contiguous 64-float range for A (its row) and B (its cluster); one b128
    // load feeds two WMMAs via even-aligned .xy/.zw halves (no repacking).
    v8f cosv = {};
    const float* ar = tl + lcol * TSTRIDE + 64 * h2;
    const float* br = mn_lds + lcol * TSTRIDE + 64 * h2;
#pragma unroll
    for (int q = 0; q < 16; ++q) {
      v4f a4 = *(const v4f*)(ar + 4 * q);
      v4f b4 = *(const v4f*)(br + 4 * q);
      cosv = __builtin_amdgcn_wmma_f32_16x16x4_f32(
          false, a4.xy, false, b4.xy, (short)0, cosv, false, false);
      cosv = __builtin_amdgcn_wmma_f32_16x16x4_f32(
          false, a4.zw, false, b4.zw, (short)0, cosv, false, false);
    }

    // ---- softmax over clusters (row j+8h lives in one 16-lane half) ------
    // |x| <= TEMP so exp cannot overflow: max-subtraction is omitted
    // (identical to reference up to fp rounding). __expf = v_exp_f32,
    // <=1 ulp on this range.
    float rr[8];
#pragma unroll
    for (int j = 0; j < 8; ++j) {
      float rsj = h2 ? rb[j + 8] : rb[j];    // two s_loads + cndmask
      float x = -TEMP * (cosv[j] * rsj);     // dn correction per row
      float e = __expf(x);
      float s = sum16(e);
      rr[j] = e / s;
    }

    if (write_r) {
#pragma unroll
      for (int j = 0; j < 8; ++j)
        r_out[(trow + j + 8 * h2) * KK + lcol] = rr[j];
    }
#pragma unroll
    for (int j = 0; j < 8; ++j) crv += rr[j];

    // ---- GEMM2: cluster_mean += r^T @ data_tile --------------------------
    // Slot map: row(c, s) = 2*c + (s&1) + 8*(s>>1). With the C-layout of r
    // (lane holds rows 8h..8h+7 of its cluster column), the A'-fragment for
    // instruction c is exactly (rr[2c], rr[2c+1]) in-lane: zero shuffles.
    // B' pairs are adjacent rows -> single ds_load_2addr, conflict-free.
#pragma unroll
    for (int c = 0; c < 4; ++c) {
      v2f af;
      af[0] = rr[2 * c];
      af[1] = rr[2 * c + 1];
      const float* b0 = tl + (2 * c + 8 * h2) * TSTRIDE + lcol;
#pragma unroll
      for (int d0 = 0; d0 < 8; ++d0) {
        v2f bf;
        bf[0] = b0[d0 * 16];
        bf[1] = b0[d0 * 16 + TSTRIDE];
        acc[d0] = __builtin_amdgcn_wmma_f32_16x16x4_f32(
            false, af, false, bf, (short)0, acc[d0], false, false);
      }
    }
  }

  // ---- epilogue: wave regs -> LDS (ds_add_f32) -> global atomics ---------
  crv += __shfl_xor(crv, 16, 32);
  if (lane < KK) atomicAdd(&cr_lds[lane], crv);
#pragma unroll
  for (int d0 = 0; d0 < 8; ++d0)
#pragma unroll
    for (int jj = 0; jj < 8; ++jj)
      atomicAdd(&cm_lds[(jj + 8 * h2) * DD + d0 * 16 + lcol], acc[d0][jj]);
  __syncthreads();
  for (int i = tid; i < KK * DD; i += 128) atomicAdd(&cmean[i], cm_lds[i]);
  if (tid < KK) atomicAdd(&cr[tid], cr_lds[tid]);
}

// --------------------------------------------------------------------------
// Kernel 4: mu = cluster_mean / cluster_r ; optionally emit final mu.
// --------------------------------------------------------------------------
__global__ void muupdate_kernel(const float* __restrict__ cmean,
                                const float* __restrict__ cr,
                                float* __restrict__ mu,
                                float* __restrict__ out_mu, int write_out) {
  int i = blockIdx.x * blockDim.x + threadIdx.x;
  if (i < KK * DD) {
    float v = cmean[i] / cr[i >> 7];
    mu[i] = v;
    if (write_out) out_mu[i] = v;
  }
}

extern "C" void kernel_launch(void* const* d_in, const int* in_sizes, int n_in,
                              void* d_out, int out_size, void* d_ws, size_t ws_size,
                              hipStream_t stream) {
  const float* emb     = (const float*)d_in[0];   // [N, 128] fp32
  const float* mu_init = (const float*)d_in[1];   // [16, 128] fp32
  float* out = (float*)d_out;                     // [2048] mu ++ [N*16] r

  // workspace layout (~4 MB)
  float* s1    = (float*)d_ws;      // [N]
  float* rsc   = s1 + NN;           // [N]
  float* mu    = rsc + NN;          // [2048]
  float* mn    = mu + KK * DD;      // [2048]
  float* cmean = mn + KK * DD;      // [2048]
  float* cr    = cmean + KK * DD;   // [16]

  rownorm_kernel<<<(NN + 7) / 8, 256, 0, stream>>>(emb, s1, rsc);

  for (int it = 0; it < NITER; ++it) {
    const float* musrc = (it == 0) ? mu_init : mu;
    const int last = (it == NITER - 1) ? 1 : 0;
    muprep_kernel<<<1, 256, 0, stream>>>(musrc, mn, cmean, cr);
    cluster_main<<<1024, 128, 0, stream>>>(emb, s1, rsc, mn, cmean, cr,
                                           out + KK * DD, last);
    muupdate_kernel<<<(KK * DD) / 256, 256, 0, stream>>>(cmean, cr, mu, out, last);
  }
}